// CellSpatialNetCaptum_77403900609090
// MI455X (gfx1250) — compile-verified
//
#include <hip/hip_runtime.h>
#include <hip/hip_bf16.h>

typedef __attribute__((ext_vector_type(2))) float v2f;
typedef __attribute__((ext_vector_type(8))) float v8f;

#define NNODES 10000
#define NEDGES 320000
#define NTYPE  36

// ---------------------------------------------------------------- utilities
__global__ void zero_kernel(float* p, int n) {
  int i = blockIdx.x * blockDim.x + threadIdx.x;
  if (i < n) p[i] = 0.0f;
}

__global__ void count_kernel(const int* __restrict__ dst, float* __restrict__ cnt, int nE) {
  int e = blockIdx.x * blockDim.x + threadIdx.x;
  if (e < nE) atomicAdd(&cnt[dst[e]], 1.0f);
}

// -------------------------------------------------------- per-edge messages
// Each work-item owns (edge, lane-slice of OUT). ew is NEVER materialized:
// each element is rebuilt in registers from LDS-staged tables.
template<int IN, int OUT, int LANES, bool STAGE_EMB>
__global__ void edge_kernel(const float* __restrict__ hin,
                            const int* __restrict__ src,
                            const int* __restrict__ dst,
                            const int* __restrict__ ety,
                            const float* __restrict__ ef,
                            const float* __restrict__ emb,
                            const float* __restrict__ wh,
                            const float* __restrict__ bh,
                            const float* __restrict__ wg,
                            const float* __restrict__ bg,
                            float* __restrict__ agg, int nE) {
  constexpr int D = IN * OUT;
  constexpr int OPL = OUT / LANES;
  extern __shared__ float smem[];
  float* s_wh = smem;              // 2*D
  float* s_bh = s_wh + 2 * D;      // D
  float* s_wg = s_bh + D;          // 2*D
  float* s_bg = s_wg + 2 * D;      // D
  float* s_emb = s_bg + D;         // NTYPE*D (optional)
  for (int i = threadIdx.x; i < 2 * D; i += blockDim.x) { s_wh[i] = wh[i]; s_wg[i] = wg[i]; }
  for (int i = threadIdx.x; i < D; i += blockDim.x)     { s_bh[i] = bh[i]; s_bg[i] = bg[i]; }
  if (STAGE_EMB)
    for (int i = threadIdx.x; i < NTYPE * D; i += blockDim.x) s_emb[i] = emb[i];
  __syncthreads();

  int stride = gridDim.x * blockDim.x;
  for (int idx = blockIdx.x * blockDim.x + threadIdx.x; idx < nE * LANES; idx += stride) {
    int e = idx / LANES;
    int sl = idx % LANES;
    int s = src[e], d = dst[e], t = ety[e];
    float e0 = ef[2 * e], e1 = ef[2 * e + 1];
    const float* er = STAGE_EMB ? (s_emb + t * D) : (emb + t * D);
    int obase = sl * OPL;
    float acc[OPL];
#pragma unroll
    for (int o = 0; o < OPL; ++o) acc[o] = 0.0f;
#pragma unroll
    for (int i = 0; i < IN; ++i) {
      float xv = hin[s * IN + i];
#pragma unroll
      for (int o = 0; o < OPL; ++o) {
        int k = i * OUT + obase + o;
        float hv = fmaf(e1, s_wh[2 * k + 1], fmaf(e0, s_wh[2 * k], s_bh[k]));
        float gv = fmaf(e1, s_wg[2 * k + 1], fmaf(e0, s_wg[2 * k], s_bg[k]));
        float ew = fmaxf(fmaf(er[k], hv, gv), 0.0f);
        acc[o] = fmaf(xv, ew, acc[o]);
      }
    }
#pragma unroll
    for (int o = 0; o < OPL; ++o)
      atomicAdd(&agg[d * OUT + obase + o], acc[o]);
  }
}

// ------------------------------------------------ node update: WMMA fp32 GEMM
// hout[m,n] = relu(agg[m,n]/max(cnt,1) + sum_k hin[m,k]*root[n,k] + bias[n])
// A(16x4 fp32): lane L holds row m=L%16, K = k0 + 2*(L/16) + {0,1}  (v2f)
// B(4x16 fp32): lane L holds col n=L%16, same K striping              (v2f)
// C/D(16x16):   VGPR r, lane L -> m = r + 8*(L/16), n = L%16          (v8f)
template<int IN, int OUT>
__global__ void node_wmma_kernel(const float* __restrict__ hin,
                                 const float* __restrict__ agg,
                                 const float* __restrict__ cnt,
                                 const float* __restrict__ root,
                                 const float* __restrict__ bias,
                                 float* __restrict__ hout, int nNodes) {
  int wave = threadIdx.x >> 5;
  int lane = threadIdx.x & 31;
  int tile = blockIdx.x * (blockDim.x >> 5) + wave;   // wave-uniform
  int nTiles = nNodes / 16;                           // N % 16 == 0
  if (tile >= nTiles) return;                         // uniform per wave -> EXEC stays full

  int m0 = tile * 16;
  int mrow = m0 + (lane & 15);
  int khalf = (lane >> 4) * 2;
  int ncol = lane & 15;

#pragma unroll
  for (int nt = 0; nt < OUT; nt += 16) {
    v8f c = {};
#pragma unroll
    for (int k0 = 0; k0 < IN; k0 += 4) {
      v2f a, b;
      a.x = hin[mrow * IN + k0 + khalf];
      a.y = hin[mrow * IN + k0 + khalf + 1];
      int n = nt + ncol;
      if (n < OUT) {
        b.x = root[n * IN + k0 + khalf];
        b.y = root[n * IN + k0 + khalf + 1];
      } else {
        b.x = 0.0f; b.y = 0.0f;
      }
      c = __builtin_amdgcn_wmma_f32_16x16x4_f32(
          /*neg_a=*/false, a, /*neg_b=*/false, b,
          /*c_mod=*/(short)0, c, /*reuse_a=*/false, /*reuse_b=*/false);
    }
    int n = nt + ncol;
    if (n < OUT) {
      float bv = bias[n];
#pragma unroll
      for (int r = 0; r < 8; ++r) {
        int m = m0 + r + 8 * (lane >> 4);
        float inv = 1.0f / fmaxf(cnt[m], 1.0f);
        float v = fmaf(agg[m * OUT + n], inv, c[r] + bv);
        hout[m * OUT + n] = fmaxf(v, 0.0f);
      }
    }
  }
}

__global__ void mean64_kernel(const float* __restrict__ h4, float* __restrict__ out, int n) {
  int i = blockIdx.x * blockDim.x + threadIdx.x;
  if (i < n) {
    float s = 0.0f;
#pragma unroll
    for (int j = 0; j < 64; ++j) s += h4[i * 64 + j];
    out[i] = s * (1.0f / 64.0f);
  }
}

// ----------------------------------------------------------------- launcher
extern "C" void kernel_launch(void* const* d_in, const int* in_sizes, int n_in,
                              void* d_out, int out_size, void* d_ws, size_t ws_size,
                              hipStream_t stream) {
  const float* x   = (const float*)d_in[0];
  const int*   src = (const int*)d_in[1];
  const int*   dst = (const int*)d_in[2];
  const int*   ety = (const int*)d_in[3];
  const float* ef  = (const float*)d_in[4];
  // per-layer params start at 5, 7 tensors each: emb, wh, bh, wg, bg, root, bias
  const float* P[4][7];
  for (int l = 0; l < 4; ++l)
    for (int j = 0; j < 7; ++j)
      P[l][j] = (const float*)d_in[5 + l * 7 + j];

  const int N = NNODES, E = NEDGES;
  float* ws  = (float*)d_ws;
  float* cnt = ws;                         // N
  float* agg = cnt + N;                    // N*64 (reused per layer)
  float* h1  = agg + (size_t)N * 64;       // N*8
  float* h2  = h1 + (size_t)N * 8;         // N*8
  float* h3  = h2 + (size_t)N * 8;         // N*8
  float* h4  = h3 + (size_t)N * 8;         // N*64

  const int TB = 256;
  // degree counts (dst identical for all 4 layers -> computed once)
  zero_kernel<<<(N + TB - 1) / TB, TB, 0, stream>>>(cnt, N);
  count_kernel<<<(E + TB - 1) / TB, TB, 0, stream>>>(dst, cnt, E);

  const int nodeGrid = (N / 16 + 7) / 8;   // 8 waves of 32 per block

  // ---- layer 1: IN=16, OUT=8, D=128 (emb staged: 42*128*4 = 21 KB LDS)
  {
    constexpr int D = 16 * 8;
    size_t shm = (size_t)(6 * D + NTYPE * D) * sizeof(float);
    zero_kernel<<<(N * 8 + TB - 1) / TB, TB, 0, stream>>>(agg, N * 8);
    edge_kernel<16, 8, 1, true><<<1024, TB, shm, stream>>>(
        x, src, dst, ety, ef, P[0][0], P[0][1], P[0][2], P[0][3], P[0][4], agg, E);
    node_wmma_kernel<16, 8><<<nodeGrid, TB, 0, stream>>>(
        x, agg, cnt, P[0][5], P[0][6], h1, N);
  }
  // ---- layers 2,3: IN=8, OUT=8, D=64 (emb staged: 42*64*4 = 10.5 KB LDS)
  {
    constexpr int D = 8 * 8;
    size_t shm = (size_t)(6 * D + NTYPE * D) * sizeof(float);
    zero_kernel<<<(N * 8 + TB - 1) / TB, TB, 0, stream>>>(agg, N * 8);
    edge_kernel<8, 8, 1, true><<<1024, TB, shm, stream>>>(
        h1, src, dst, ety, ef, P[1][0], P[1][1], P[1][2], P[1][3], P[1][4], agg, E);
    node_wmma_kernel<8, 8><<<nodeGrid, TB, 0, stream>>>(
        h1, agg, cnt, P[1][5], P[1][6], h2, N);

    zero_kernel<<<(N * 8 + TB - 1) / TB, TB, 0, stream>>>(agg, N * 8);
    edge_kernel<8, 8, 1, true><<<1024, TB, shm, stream>>>(
        h2, src, dst, ety, ef, P[2][0], P[2][1], P[2][2], P[2][3], P[2][4], agg, E);
    node_wmma_kernel<8, 8><<<nodeGrid, TB, 0, stream>>>(
        h2, agg, cnt, P[2][5], P[2][6], h3, N);
  }
  // ---- layer 4: IN=8, OUT=64, D=512; 4 lanes/edge, 16 outs each.
  // emb (73.7 KB) stays in L2; only 12 KB of wh/bh/wg/bg staged in LDS.
  {
    constexpr int D = 8 * 64;
    size_t shm = (size_t)(6 * D) * sizeof(float);
    zero_kernel<<<(N * 64 + TB - 1) / TB, TB, 0, stream>>>(agg, N * 64);
    edge_kernel<8, 64, 4, false><<<1024, TB, shm, stream>>>(
        h3, src, dst, ety, ef, P[3][0], P[3][1], P[3][2], P[3][3], P[3][4], agg, E);
    node_wmma_kernel<8, 64><<<nodeGrid, TB, 0, stream>>>(
        h3, agg, cnt, P[3][5], P[3][6], h4, N);
  }
  // ---- final channel mean -> d_out [N] fp32
  mean64_kernel<<<(N + TB - 1) / TB, TB, 0, stream>>>(h4, (float*)d_out, N);
}